// ManoHand_43284680409172
// MI455X (gfx1250) — compile-verified
//
#include <hip/hip_runtime.h>
#include <math.h>

typedef __attribute__((ext_vector_type(2))) float v2f;
typedef __attribute__((ext_vector_type(8))) float v8f;

#define BATCH 4096
#define NVERT 778
#define NJOINT 16
#define NBPOSE 135
#define NCOL 2334   // NVERT*3
#define NSUB 4      // N-subtiles (16 cols each) per wave

// ---------------------------------------------------------------------------
// Kernel 1: pose PCA + Rodrigues -> Rs[B][16][9], pose_feature[B][135]
// ---------------------------------------------------------------------------
__global__ __launch_bounds__(256) void prep_kernel(
    const float* __restrict__ theta,        // [B,48]
    const float* __restrict__ hands_mean,   // [45]
    const float* __restrict__ hands_comp,   // [45,45]
    float* __restrict__ Rs,                 // [B,16,9]
    float* __restrict__ pf)                 // [B,135]
{
    int b = blockIdx.x * blockDim.x + threadIdx.x;
    if (b >= BATCH) return;
    const float* th = theta + b * 48;

    float pose[48];
    pose[0] = th[0]; pose[1] = th[1]; pose[2] = th[2];
    for (int i = 0; i < 45; ++i) {
        float s = hands_mean[i];
        for (int p = 0; p < 45; ++p) s += th[3 + p] * hands_comp[p * 45 + i];
        pose[3 + i] = s;
    }

    for (int j = 0; j < NJOINT; ++j) {
        float rx = pose[3 * j + 0], ry = pose[3 * j + 1], rz = pose[3 * j + 2];
        float ex = rx + 1e-8f, ey = ry + 1e-8f, ez = rz + 1e-8f;
        float ang = sqrtf(ex * ex + ey * ey + ez * ez);
        float inv = 1.0f / ang;
        float x = rx * inv, y = ry * inv, z = rz * inv;
        float sn = sinf(ang), cs = cosf(ang), t = 1.0f - cs;
        float R[9];
        R[0] = 1.0f - t * (y * y + z * z);
        R[1] = sn * (-z) + t * (x * y);
        R[2] = sn * ( y) + t * (x * z);
        R[3] = sn * ( z) + t * (x * y);
        R[4] = 1.0f - t * (x * x + z * z);
        R[5] = sn * (-x) + t * (y * z);
        R[6] = sn * (-y) + t * (x * z);
        R[7] = sn * ( x) + t * (y * z);
        R[8] = 1.0f - t * (x * x + y * y);
        float* Rd = Rs + (size_t)b * 144 + j * 9;
        for (int e = 0; e < 9; ++e) Rd[e] = R[e];
        if (j >= 1) {
            float* pd = pf + (size_t)b * NBPOSE + (j - 1) * 9;
            for (int e = 0; e < 9; ++e)
                pd[e] = R[e] - ((e == 0 || e == 4 || e == 8) ? 1.0f : 0.0f);
        }
    }
}

// ---------------------------------------------------------------------------
// Kernel 2/4: fp32 WMMA GEMM  Out[M,N] = A[M,K] @ W[K,N] (+ addVec[N]) (+ addMat[M,N])
// One wave computes a 16x64 C tile (4 accumulators sharing one A fragment).
// All global accesses are base + 32-bit element offset so the backend emits
// global_load/store (SADDR form), never flat. Main K-loop is mask-free:
// N-edges handled by offset clamping only (garbage columns never stored);
// the K tail zero-masks the A fragment only (0 * clamped-B == 0).
// EXEC stays all-ones through every WMMA.
// ---------------------------------------------------------------------------
__global__ __launch_bounds__(256) void wmma_gemm_kernel(
    const float* __restrict__ A, int lda,
    const float* __restrict__ W,          // [K,N]
    const float* __restrict__ addVec,     // [N] or null
    const float* __restrict__ addMat,     // [M,N] or null
    float* __restrict__ Out,
    int M, int N, int K)
{
    int wave = blockIdx.x * (blockDim.x >> 5) + (threadIdx.x >> 5);
    int lane = threadIdx.x & 31;
    int ntiles  = (N + 15) >> 4;
    int ngroups = (ntiles + NSUB - 1) / NSUB;
    int mtiles  = M >> 4;                  // M divisible by 16
    if (wave >= mtiles * ngroups) return;

    int mt = wave / ngroups, gt = wave - mt * ngroups;
    int m0 = mt << 4;
    int g0 = gt * (NSUB << 4);             // first column of this wave's group

    int lan16 = lane & 15;
    int koff  = (lane >> 4) << 1;          // 0 or 2: K pair owned by this half-wave
    int mrow  = m0 + lan16;                // A-frag row

    // Per-subtile clamped column offsets (32-bit, clamp-only)
    int ncl[NSUB];
#pragma unroll
    for (int t = 0; t < NSUB; ++t) {
        int nc = g0 + (t << 4) + lan16;
        ncl[t] = (nc < N) ? nc : (N - 1);
    }

    int woff[NSUB];
#pragma unroll
    for (int t = 0; t < NSUB; ++t) woff[t] = koff * N + ncl[t];
    int aoff = mrow * lda + koff;
    int stepW = 4 * N;

    v8f acc[NSUB];
#pragma unroll
    for (int t = 0; t < NSUB; ++t) acc[t] = (v8f){0.f,0.f,0.f,0.f,0.f,0.f,0.f,0.f};

    int kfull = K & ~3;
    for (int kb = 0; kb < kfull; kb += 4) {
        v2f af;
        af[0] = A[aoff];
        af[1] = A[aoff + 1];
#pragma unroll
        for (int t = 0; t < NSUB; ++t) {
            v2f bf;
            bf[0] = W[woff[t]];
            bf[1] = W[woff[t] + N];
            acc[t] = __builtin_amdgcn_wmma_f32_16x16x4_f32(
                false, af, false, bf, (short)0, acc[t], false, false);
            woff[t] += stepW;
        }
        aoff += 4;
    }
    if (K & 3) {                            // masked K tail (A-frag only)
        int k0 = kfull + koff, k1 = k0 + 1;
        float km0 = (k0 < K) ? 1.0f : 0.0f;
        float km1 = (k1 < K) ? 1.0f : 0.0f;
        int k0c = (k0 < K) ? k0 : 0;
        int k1c = (k1 < K) ? k1 : 0;
        v2f af;
        af[0] = A[mrow * lda + k0c] * km0;
        af[1] = A[mrow * lda + k1c] * km1;
        int r0 = k0c * N, r1 = k1c * N;
#pragma unroll
        for (int t = 0; t < NSUB; ++t) {
            v2f bf;
            bf[0] = W[r0 + ncl[t]];
            bf[1] = W[r1 + ncl[t]];
            acc[t] = __builtin_amdgcn_wmma_f32_16x16x4_f32(
                false, af, false, bf, (short)0, acc[t], false, false);
        }
    }

    // Store: C layout -> VGPR r: lanes 0-15 row m0+r, lanes 16-31 row m0+r+8
    int mbase = m0 + ((lane >> 4) << 3);
#pragma unroll
    for (int t = 0; t < NSUB; ++t) {
        int n = g0 + (t << 4) + lan16;
        if (n < N) {
            float av = addVec ? addVec[n] : 0.0f;
            int o = mbase * N + n;
#pragma unroll
            for (int r = 0; r < 8; ++r) {
                float v = acc[t][r] + av;
                if (addMat) v += addMat[o];
                Out[o] = v;
                o += N;
            }
        }
    }
}

// ---------------------------------------------------------------------------
// Kernel 3: J[b,j,c] = sum_v v_shaped[b, 3v+c] * Jreg[v,j]
// ---------------------------------------------------------------------------
__global__ __launch_bounds__(256) void jreg_kernel(
    const float* __restrict__ vsh,        // [B, NCOL]
    const float* __restrict__ Jr,         // [NVERT, 16]
    float* __restrict__ Jout)             // [B, 16, 3]
{
    int t = blockIdx.x * blockDim.x + threadIdx.x;
    if (t >= BATCH * 48) return;
    int b = t / 48, jc = t - b * 48;
    int j = jc / 3, c = jc - j * 3;
    const float* vb = vsh + (size_t)b * NCOL;
    float s = 0.0f;
    for (int v = 0; v < NVERT; ++v) s += vb[v * 3 + c] * Jr[v * 16 + j];
    Jout[(size_t)b * 48 + j * 3 + c] = s;
}

// ---------------------------------------------------------------------------
// Kernel 5: kinematic chain -> relative skinning transforms A[b][16][4][4]
// ---------------------------------------------------------------------------
__device__ inline void make_local(const float* R, float tx, float ty, float tz, float* L) {
    L[0]=R[0]; L[1]=R[1]; L[2]=R[2];  L[3]=tx;
    L[4]=R[3]; L[5]=R[4]; L[6]=R[5];  L[7]=ty;
    L[8]=R[6]; L[9]=R[7]; L[10]=R[8]; L[11]=tz;
    L[12]=0.f; L[13]=0.f; L[14]=0.f;  L[15]=1.f;
}
__device__ inline void mat4_mul(const float* A, const float* B, float* C) {
    for (int p = 0; p < 4; ++p)
        for (int q = 0; q < 4; ++q)
            C[p*4+q] = A[p*4+0]*B[0*4+q] + A[p*4+1]*B[1*4+q]
                     + A[p*4+2]*B[2*4+q] + A[p*4+3]*B[3*4+q];
}
__device__ inline void store_adj(float* dst, const float* res, float jx, float jy, float jz) {
    for (int p = 0; p < 4; ++p) {
        float ib = res[p*4+0]*jx + res[p*4+1]*jy + res[p*4+2]*jz;
        dst[p*4+0] = res[p*4+0];
        dst[p*4+1] = res[p*4+1];
        dst[p*4+2] = res[p*4+2];
        dst[p*4+3] = res[p*4+3] - ib;
    }
}

__global__ __launch_bounds__(256) void chain_kernel(
    const float* __restrict__ Rs,   // [B,16,9]
    const float* __restrict__ Jb,   // [B,16,3]
    float* __restrict__ Abuf)       // [B,16,16]
{
    int b = blockIdx.x * blockDim.x + threadIdx.x;
    if (b >= BATCH) return;
    const float* R = Rs + (size_t)b * 144;
    const float* J = Jb + (size_t)b * 48;
    float* Ab = Abuf + (size_t)b * 256;

    float root[16];
    make_local(R, J[0], J[1], J[2], root);
    store_adj(Ab, root, J[0], J[1], J[2]);

    for (int c = 0; c < 5; ++c) {
        float prev[16];
        for (int e = 0; e < 16; ++e) prev[e] = root[e];
        int par = 0;
        for (int s = 0; s < 3; ++s) {
            int j = c * 3 + 1 + s;           // joints 1..15, depth-3 chains
            float L[16], cur[16];
            make_local(R + j * 9,
                       J[3*j+0] - J[3*par+0],
                       J[3*j+1] - J[3*par+1],
                       J[3*j+2] - J[3*par+2], L);
            mat4_mul(prev, L, cur);
            store_adj(Ab + j * 16, cur, J[3*j+0], J[3*j+1], J[3*j+2]);
            for (int e = 0; e < 16; ++e) prev[e] = cur[e];
            par = j;
        }
    }
}

// ---------------------------------------------------------------------------
// Kernel 6: LBS skinning + joint regression + finger joints. One block/batch.
// ---------------------------------------------------------------------------
__global__ __launch_bounds__(256) void skin_kernel(
    const float* __restrict__ vposed,   // [B, NCOL]
    const float* __restrict__ Abuf,     // [B,16,16]
    const float* __restrict__ wts,      // [NVERT,16]
    const float* __restrict__ Jr,       // [NVERT,16]
    float* __restrict__ out)            // [B,21,3]
{
    int b = blockIdx.x;
    int tid = threadIdx.x;
    __shared__ float Als[256];
    __shared__ float jacc[48];
    Als[tid] = Abuf[(size_t)b * 256 + tid];
    if (tid < 48) jacc[tid] = 0.0f;
    __syncthreads();

    const int fidx[5] = {734, 333, 443, 555, 678};

    for (int v = tid; v < NVERT; v += 256) {
        const float* vp = vposed + (size_t)b * NCOL + v * 3;
        float x = vp[0], y = vp[1], z = vp[2];

        float Mr[12];
        for (int e = 0; e < 12; ++e) Mr[e] = 0.0f;
        for (int j = 0; j < NJOINT; ++j) {
            float wj = wts[v * 16 + j];
            const float* Aj = &Als[j * 16];
            for (int p = 0; p < 3; ++p)
                for (int q = 0; q < 4; ++q)
                    Mr[p*4+q] += wj * Aj[p*4+q];
        }
        float vx = Mr[0]*x + Mr[1]*y + Mr[2]*z  + Mr[3];
        float vy = Mr[4]*x + Mr[5]*y + Mr[6]*z  + Mr[7];
        float vz = Mr[8]*x + Mr[9]*y + Mr[10]*z + Mr[11];

        for (int jj = 0; jj < NJOINT; ++jj) {
            float rw = Jr[v * 16 + jj];
            atomicAdd(&jacc[jj*3+0], rw * vx);
            atomicAdd(&jacc[jj*3+1], rw * vy);
            atomicAdd(&jacc[jj*3+2], rw * vz);
        }
        for (int f = 0; f < 5; ++f) {
            if (v == fidx[f]) {
                float* fo = out + (size_t)b * 63 + (16 + f) * 3;
                fo[0] = vx; fo[1] = vy; fo[2] = vz;
            }
        }
    }
    __syncthreads();
    if (tid < 48) out[(size_t)b * 63 + tid] = jacc[tid];
}

// ---------------------------------------------------------------------------
extern "C" void kernel_launch(void* const* d_in, const int* in_sizes, int n_in,
                              void* d_out, int out_size, void* d_ws, size_t ws_size,
                              hipStream_t stream) {
    const float* beta        = (const float*)d_in[0];
    const float* theta       = (const float*)d_in[1];
    const float* v_template  = (const float*)d_in[2];
    const float* shapedirs   = (const float*)d_in[3];
    const float* posedirs    = (const float*)d_in[4];
    const float* J_regressor = (const float*)d_in[5];
    const float* weights     = (const float*)d_in[6];
    const float* hands_mean  = (const float*)d_in[7];
    const float* hands_comp  = (const float*)d_in[8];
    float* out = (float*)d_out;

    float* w = (float*)d_ws;
    float* vshaped = w;  w += (size_t)BATCH * NCOL;    // 38.2 MB
    float* vposed  = w;  w += (size_t)BATCH * NCOL;    // 38.2 MB
    float* Rs      = w;  w += (size_t)BATCH * 144;
    float* pf      = w;  w += (size_t)BATCH * NBPOSE;
    float* Jb      = w;  w += (size_t)BATCH * 48;
    float* Ab      = w;  w += (size_t)BATCH * 256;     // total ~86 MB

    prep_kernel<<<BATCH / 256, 256, 0, stream>>>(theta, hands_mean, hands_comp, Rs, pf);

    int ntiles  = (NCOL + 15) / 16;                 // 146
    int ngroups = (ntiles + NSUB - 1) / NSUB;       // 37
    int totalWaves = (BATCH / 16) * ngroups;        // 9472
    int gemmBlocks = (totalWaves + 7) / 8;          // 8 waves / 256-thread block

    // v_shaped = beta @ shapedirs + v_template
    wmma_gemm_kernel<<<gemmBlocks, 256, 0, stream>>>(
        beta, 10, shapedirs, v_template, nullptr, vshaped, BATCH, NCOL, 10);

    // J = regress(v_shaped)
    jreg_kernel<<<(BATCH * 48 + 255) / 256, 256, 0, stream>>>(vshaped, J_regressor, Jb);

    // v_posed = pose_feature @ posedirs + v_shaped
    wmma_gemm_kernel<<<gemmBlocks, 256, 0, stream>>>(
        pf, NBPOSE, posedirs, nullptr, vshaped, vposed, BATCH, NCOL, NBPOSE);

    // kinematic chain -> skinning transforms
    chain_kernel<<<BATCH / 256, 256, 0, stream>>>(Rs, Jb, Ab);

    // skinning + joint regression + fingers
    skin_kernel<<<BATCH, 256, 0, stream>>>(vposed, Ab, weights, J_regressor, out);
}